// bicon_loss_71588514889847
// MI455X (gfx1250) — compile-verified
//
#include <hip/hip_runtime.h>
#include <hip/hip_bf16.h>

// Shapes from the reference
#define BATCH 8
#define CHANS 8
#define HH 512
#define WW 512
#define PLANE (HH * WW)          // 262144
#define IMG   (CHANS * PLANE)    // per-batch stride of 8-channel tensors
#define NPIX  (BATCH * HH * WW)  // 2,097,152
#define NEL   (BATCH * CHANS * HH * WW)  // 16,777,216

typedef __attribute__((ext_vector_type(2))) float v2f;
typedef __attribute__((ext_vector_type(8))) float v8f;

// ---------------------------------------------------------------------------
// Fused pass: sigmoid + bilateral voting (8-neighborhood stencil) + three BCE
// losses, weighted, reduced to one partial per block. One thread per pixel,
// coalesced along x; neighbor re-reads are absorbed by the 192MB L2.
// All transcendentals are single hardware ops (v_exp_f32 / v_log_f32);
// 0/1 targets let every BCE element use exactly one log via select.
// ---------------------------------------------------------------------------
__global__ __launch_bounds__(256) void bicon_fused_kernel(
    const float* __restrict__ out2, const float* __restrict__ out3,
    const float* __restrict__ out4, const float* __restrict__ out5,
    const float* __restrict__ target, const float* __restrict__ con,
    float* __restrict__ partial) {
  const int idx = blockIdx.x * 256 + threadIdx.x;   // pixel id, exact multiple
  const int x = idx & (WW - 1);
  const int y = (idx >> 9) & (HH - 1);
  const int b = idx >> 18;
  const int base8 = b * IMG + y * WW + x;           // channel ch at +ch*PLANE

  // connectivity target (8 channels) -> edge mask
  float ct[8];
  float sc = 0.f;
#pragma unroll
  for (int ch = 0; ch < 8; ++ch) {
    ct[ch] = con[base8 + ch * PLANE];
    sc += ct[ch];
  }
  const float edge = (sc > 0.5f && sc < 7.5f) ? 1.0f : 0.0f;
  const bool  tpos = target[b * PLANE + y * WW + x] > 0.5f;

  // vote channel i: center channel i, neighbor channel 7-i at (y+dy, x+dx)
  const int dy[8] = {-1, -1, -1, 0, 0, 1, 1, 1};
  const int dx[8] = {-1, 0, 1, -1, 1, -1, 0, 1};

  const float* outs[4] = {out2, out3, out4, out5};
  const float dw_[4] = {1.0f, 0.8f, 0.6f, 0.4f};
  const float cw_[4] = {0.8f, 0.64f, 0.48f, 0.32f};
  const float bw_[4] = {0.2f, 0.16f, 0.12f, 0.08f};
  const float invNpix = 1.0f / (float)NPIX;
  const float invNel  = 1.0f / (float)NEL;

  float acc = 0.f;
#pragma unroll
  for (int k = 0; k < 4; ++k) {
    const float* __restrict__ p = outs[k];
    // warm next row into L2 (gfx1250 global_prefetch_b8)
    __builtin_prefetch(p + base8 + WW, 0, 1);

    // Center channels: sigmoid value + connectivity BCE term.
    //   sigma = num/(1+e), 1-sigma = num'/(1+e), num in {1, e}, log(e) = -|v|
    //   => BCE term = ((v>=0)==ct ? 0 : -|v|) - log(1+e), one v_log total.
    float o[8];
    float sum_con = 0.f;
#pragma unroll
    for (int ch = 0; ch < 8; ++ch) {
      const float v  = p[base8 + ch * PLANE];
      const float av = fabsf(v);
      const float e  = __expf(-av);           // exp(-|v|)
      const float l  = __logf(1.0f + e);      // softplus(-|v|)
      const float inv = 1.0f / (1.0f + e);
      o[ch] = (v >= 0.f ? 1.0f : e) * inv;    // sigmoid(v)
      const bool ctb = ct[ch] > 0.5f;
      const float numlog = ((v >= 0.f) == ctb) ? 0.0f : -av;
      sum_con += fmaxf(numlog - l, -100.f);
    }

    // Shifted-neighbor sigmoids -> vote products (zero-padded at borders)
    float a[8];
#pragma unroll
    for (int i = 0; i < 8; ++i) {
      const int yy = y + dy[i];
      const int xx = x + dx[i];
      float nb = 0.f;
      if (yy >= 0 && yy < HH && xx >= 0 && xx < WW) {
        const float v = p[b * IMG + (7 - i) * PLANE + yy * WW + xx];
        nb = 1.0f / (1.0f + __expf(-v));
      }
      a[i] = o[i] * nb;
    }

    // Bilateral-vote BCE: one log per channel via 0/1-target select
    float s = 0.f, vmin = a[0], sum_bi = 0.f;
#pragma unroll
    for (int i = 0; i < 8; ++i) {
      s += a[i];
      vmin = fminf(vmin, a[i]);
      const float sel = (ct[i] > 0.5f) ? a[i] : (1.0f - a[i]);
      sum_bi += fmaxf(__logf(sel), -100.f);   // log(0) -> -inf -> clamp
    }

    // Edge-decoupled map BCE (one log)
    const float glo = s * 0.125f;
    const float pm  = (1.f - vmin) * edge;
    const float d   = glo * (1.f - edge) + pm;
    const float seld = tpos ? d : (1.0f - d);
    const float term_d = fmaxf(__logf(seld), -100.f);

    // BCE is -mean(...): accumulate negated, pre-scaled contributions
    acc -= dw_[k] * term_d * invNpix + (cw_[k] * sum_con + bw_[k] * sum_bi) * invNel;
  }

  // deterministic block reduction
  __shared__ float red[256];
  red[threadIdx.x] = acc;
  __syncthreads();
#pragma unroll
  for (int off = 128; off > 0; off >>= 1) {
    if (threadIdx.x < off) red[threadIdx.x] += red[threadIdx.x + off];
    __syncthreads();
  }
  if (threadIdx.x == 0) partial[blockIdx.x] = red[0];
}

// ---------------------------------------------------------------------------
// Final reduction on a single wave32 (EXEC all ones, as WMMA requires).
// 64 strided f32 accumulators form a 16x4 A matrix; B = ones, so
// D = A*B holds row-sums; column 0 (lanes 0 and 16) sums to the exact total.
// Multiplies are by 1.0 -> exact f32 result, deterministic order per lane.
// ---------------------------------------------------------------------------
__global__ __launch_bounds__(32) void bicon_wmma_reduce_kernel(
    const float* __restrict__ partial, float* __restrict__ out, int n) {
  const int lane = threadIdx.x;
  float s0 = 0.f, s1 = 0.f;
  for (int i = lane; i < n; i += 64) s0 += partial[i];
  for (int i = lane + 32; i < n; i += 64) s1 += partial[i];

  v2f a;
  a.x = s0;
  a.y = s1;
  v2f bones;
  bones.x = 1.0f;
  bones.y = 1.0f;
  v8f c = {};
  // D[m][n] = sum_k A[m][k] * 1.0  (row sums broadcast to all 16 columns)
  c = __builtin_amdgcn_wmma_f32_16x16x4_f32(
      /*neg_a=*/false, a, /*neg_b=*/false, bones,
      /*c_mod=*/(short)0, c, /*reuse_a=*/false, /*reuse_b=*/false);

  // lane L (L<16) holds D[0..7][L] in c[0..7]; lane L+16 holds D[8..15][L]
  float t = c[0] + c[1] + c[2] + c[3] + c[4] + c[5] + c[6] + c[7];
  float t_hi = __shfl(t, 16, 32);   // lane 0 fetches rows 8..15 of column 0
  if (lane == 0) out[0] = t + t_hi;
}

extern "C" void kernel_launch(void* const* d_in, const int* in_sizes, int n_in,
                              void* d_out, int out_size, void* d_ws, size_t ws_size,
                              hipStream_t stream) {
  (void)in_sizes; (void)n_in; (void)out_size; (void)ws_size;
  const float* out2 = (const float*)d_in[0];
  const float* out3 = (const float*)d_in[1];
  const float* out4 = (const float*)d_in[2];
  const float* out5 = (const float*)d_in[3];
  const float* target = (const float*)d_in[4];
  const float* con = (const float*)d_in[5];
  float* partial = (float*)d_ws;              // 8192 floats = 32 KB scratch

  const int nblocks = NPIX / 256;             // 8192, exact
  bicon_fused_kernel<<<nblocks, 256, 0, stream>>>(out2, out3, out4, out5,
                                                  target, con, partial);
  bicon_wmma_reduce_kernel<<<1, 32, 0, stream>>>(partial, (float*)d_out, nblocks);
}